// WaveletDWTLayer_50371376447903
// MI455X (gfx1250) — compile-verified
//
#include <hip/hip_runtime.h>
#include <math.h>

// ---------------------------------------------------------------------------
// db4 wavelet analysis + packed outputs + fused 2-stage synthesis (MI455X)
// Memory-bound (~1.4 flop/byte): streaming kernels, vectorized global traffic,
// global_prefetch_b8 to warm the 192MB L2, and CDNA5 async global->LDS
// staging (ASYNCcnt + s_wait_asynccnt) for the reused cD window. wave32.
// ---------------------------------------------------------------------------

#define L_SIG   65536      // input signal length
#define L_CA    32771      // (L + 8 - 1) / 2
#define L_CH    131072     // 2 * L  (output-2 row: ch0 = x2, ch1 = cD+FILLER)
#define L_REC   131066     // 2*L - 6 (final reconstruction length)
#define FILLER_V 10.1f

#define AS3 __attribute__((address_space(3)))

// LDS byte offset of a generic pointer into a __shared__ object
// (addrspace(3) pointers on AMDGPU are 32-bit offsets from the LDS base).
__device__ __forceinline__ unsigned lds_off(const void* p) {
    return (unsigned)(size_t)(AS3 const void*)p;
}

// Build rec_lo / rec_hi from the raw 8-tap weights, matching the reference:
//   rec_lo = sqrt(2) * w / sum(w) ;  rec_hi[i] = (-1)^i * rec_lo[7-i]
// (dec filters fold into the correlation form used below:
//   cA[n] = sum_t xp[2n+t]*rec_lo[t],  cD[n] = sum_t xp[2n+t]*rec_hi[t])
__device__ __forceinline__ void wavelet_filters(const float* __restrict__ w,
                                                float rl[8], float rh[8]) {
    float t[8];
    float s = 0.f;
#pragma unroll
    for (int i = 0; i < 8; ++i) { t[i] = w[i]; s += t[i]; }
    const float s2 = sqrtf(2.0f);
#pragma unroll
    for (int i = 0; i < 8; ++i) rl[i] = (s2 * t[i]) / s;
#pragma unroll
    for (int i = 0; i < 8; ++i) rh[i] = (i & 1) ? -rl[7 - i] : rl[7 - i];
}

// ---------------------------------------------------------------------------
// Kernel A: analysis.  Thread p computes cA[2p..2p+1], cD[2p..2p+1] from
// xp[4p .. 4p+9] with symmetric reflection (pad 6 each side).
// cD is written straight into output-2 channel 1 (re-read by kernel C).
// ---------------------------------------------------------------------------
__global__ __launch_bounds__(256)
void wvlt_analysis(const float* __restrict__ x1, const float* __restrict__ w,
                   float* __restrict__ outA, float* __restrict__ out2) {
    const int b = blockIdx.y;
    const int p = blockIdx.x * blockDim.x + threadIdx.x;   // pair index
    if (p > (L_CA - 1) / 2) return;                        // p <= 16385

    float rl[8], rh[8];
    wavelet_filters(w, rl, rh);

    const float* xr = x1 + (size_t)b * L_SIG;

    // Warm global L2 ahead of the stream (gfx1250 global_prefetch_b8).
    {
        int pf = 4 * p + 4096;
        if (pf < L_SIG) __builtin_prefetch(xr + pf, 0, 3);
    }

    float xv[10];
#pragma unroll
    for (int j = 0; j < 10; ++j) {
        int m = 4 * p + j - 6;                       // xp[i] = sig[i-6] interior
        m = (m < 0) ? (-m - 1) : m;                  // left symmetric reflect
        m = (m >= L_SIG) ? (2 * L_SIG - 1 - m) : m;  // right symmetric reflect
        xv[j] = xr[m];
    }

    float cA0 = 0.f, cD0 = 0.f, cA1 = 0.f, cD1 = 0.f;
#pragma unroll
    for (int t = 0; t < 8; ++t) {
        cA0 = fmaf(xv[t],     rl[t], cA0);
        cD0 = fmaf(xv[t],     rh[t], cD0);
        cA1 = fmaf(xv[t + 2], rl[t], cA1);
        cD1 = fmaf(xv[t + 2], rh[t], cD1);
    }

    const int n0 = 2 * p;
    float* cArow = outA + (size_t)b * L_CA;
    float* cDrow = out2 + (size_t)b * L_CH + L_SIG;  // channel 1
    cArow[n0] = cA0;
    cDrow[n0] = cD0;
    if (n0 + 1 < L_CA) {                             // last pair has no odd half
        cArow[n0 + 1] = cA1;
        cDrow[n0 + 1] = cD1;
    }
}

// ---------------------------------------------------------------------------
// Kernel B: output-2 packing.  t < 16384: float4 copy x2 -> channel 0.
// t >= 16384: scalar FILLER fill of channel 1 positions [32771, 65536).
// ---------------------------------------------------------------------------
__global__ __launch_bounds__(256)
void wvlt_pack(const float* __restrict__ x2, float* __restrict__ out2) {
    const int b = blockIdx.y;
    const int t = blockIdx.x * blockDim.x + threadIdx.x;   // [0, 49152)
    float* row = out2 + (size_t)b * L_CH;
    if (t < L_SIG / 4) {
        const float4 v = ((const float4*)(x2 + (size_t)b * L_SIG))[t];
        ((float4*)row)[t] = v;
    } else {
        const int j = L_SIG / 2 + (t - L_SIG / 4);         // [32768, 65536)
        if (j >= L_CA) row[L_SIG + j] = FILLER_V;          // skip cD region
    }
}

// ---------------------------------------------------------------------------
// Kernel C: fused synthesis.  Both transposed convs obey
//   y[m] = sum_i in[i] * k[m - 2i + 6]
// so output pair (2t,2t+1) needs exactly in[t..t+3] (even/odd coeff subsets).
// Thread q emits rec2[4q..4q+3]:
//   rec1[2q..2q+4] recomputed in registers from cD[q..q+5]  (never hits memory)
//   + x2[2q..2q+4] window.
// The block's cD tile (256+6 floats, each element reused ~6x across threads)
// is staged into LDS with CDNA5 async global->LDS copies (ASYNCcnt).
// ---------------------------------------------------------------------------
__global__ __launch_bounds__(256)
void wvlt_synthesis(const float* __restrict__ x2, const float* __restrict__ w,
                    const float* __restrict__ out2, float* __restrict__ out3) {
    __shared__ float tile[264];                            // 256 + 6 halo (+pad)

    const int b   = blockIdx.y;
    const int tid = threadIdx.x;
    const int q0  = blockIdx.x * 256;
    const int q   = q0 + tid;

    const float* cD = out2 + (size_t)b * L_CH + L_SIG;     // channel 1
    const float* xr = x2 + (size_t)b * L_SIG;
    float*       yr = out3 + (size_t)b * L_REC;

    // L2 stream warmer (global_prefetch_b8).
    {
        int px = 2 * q + 4096;
        if (px < L_SIG) __builtin_prefetch(xr + px, 0, 3);
    }

    // ---- async stage cD[q0 .. q0+261] -> LDS (all in-bounds: padded row) ----
    {
        const float* src = cD + q0 + tid;
        unsigned     dst = lds_off(tile + tid);
        asm volatile("global_load_async_to_lds_b32 %0, %1, off"
                     :: "v"(dst), "v"(src) : "memory");
        if (tid < 6) {
            const float* src2 = cD + q0 + 256 + tid;
            unsigned     dst2 = lds_off(tile + 256 + tid);
            asm volatile("global_load_async_to_lds_b32 %0, %1, off"
                         :: "v"(dst2), "v"(src2) : "memory");
        }
        asm volatile("s_wait_asynccnt 0x0" ::: "memory");
        __syncthreads();
    }

    if (q > (L_REC - 1) / 4) return;                       // q <= 32766

    float rl[8], rh[8];
    wavelet_filters(w, rl, rh);

    // cD window from LDS (tile[tid+5] at the row tail is FILLER: unused).
    float c[6];
#pragma unroll
    for (int j = 0; j < 6; ++j) c[j] = tile[tid + j];

    // rec1[2q + r], r = 0..4 :
    //   r even: sum_j cD[q + r/2 + j]     * rh[6-2j]
    //   r odd : sum_j cD[q + (r-1)/2 + j] * rh[7-2j]
    float r1[5];
#pragma unroll
    for (int r = 0; r < 5; ++r) {
        const int base = r >> 1;
        float acc = 0.f;
        if ((r & 1) == 0) {
#pragma unroll
            for (int j = 0; j < 4; ++j) acc = fmaf(c[base + j], rh[6 - 2 * j], acc);
        } else {
#pragma unroll
            for (int j = 0; j < 4; ++j) acc = fmaf(c[base + j], rh[7 - 2 * j], acc);
        }
        r1[r] = acc;
    }

    // x2 window x2[2q .. 2q+4]; only the last index can run off the row.
    float xw[5];
#pragma unroll
    for (int j = 0; j < 5; ++j) {
        int idx = 2 * q + j;
        idx = (idx > L_SIG - 1) ? (L_SIG - 1) : idx;
        xw[j] = xr[idx];
    }

    // rec2[4q + u]: u even -> coeff set k[6-2j], u odd -> k[7-2j];
    // window offset (u>>1) into r1[] / xw[].
    float y[4];
#pragma unroll
    for (int u = 0; u < 4; ++u) {
        const int off = u >> 1;
        float acc = 0.f;
        if ((u & 1) == 0) {
#pragma unroll
            for (int j = 0; j < 4; ++j)
                acc = fmaf(r1[off + j], rl[6 - 2 * j], fmaf(xw[off + j], rh[6 - 2 * j], acc));
        } else {
#pragma unroll
            for (int j = 0; j < 4; ++j)
                acc = fmaf(r1[off + j], rl[7 - 2 * j], fmaf(xw[off + j], rh[7 - 2 * j], acc));
        }
        y[u] = acc;
    }

    // Rows are 8B-aligned (L_REC % 4 == 2), so store as float2 pairs.
    const int m0 = 4 * q;
    ((float2*)(yr + m0))[0] = make_float2(y[0], y[1]);      // m0+1 <= 131065 always
    if (m0 + 3 < L_REC)
        ((float2*)(yr + m0))[1] = make_float2(y[2], y[3]);  // tail thread drops these
}

// ---------------------------------------------------------------------------
extern "C" void kernel_launch(void* const* d_in, const int* in_sizes, int n_in,
                              void* d_out, int out_size, void* d_ws, size_t ws_size,
                              hipStream_t stream) {
    (void)n_in; (void)out_size; (void)d_ws; (void)ws_size;

    const float* x1 = (const float*)d_in[0];
    const float* x2 = (const float*)d_in[1];
    // d_in[2] (x3) is unused by the reference.
    const float* w  = (const float*)d_in[3];

    const int B = in_sizes[0] / L_SIG;        // 256

    float* outA = (float*)d_out;                              // B * 32771
    float* out2 = outA + (size_t)B * L_CA;                    // B * 2 * 65536
    float* out3 = out2 + (size_t)B * L_CH;                    // B * 131066

    const dim3 blk(256, 1, 1);
    // A: 16386 pair-threads/row -> 65 blocks of 256
    wvlt_analysis<<<dim3(65, B, 1), blk, 0, stream>>>(x1, w, outA, out2);
    // B: 16384 vec4 copies + 32768 filler slots = 49152 threads/row -> 192 blocks
    wvlt_pack<<<dim3(192, B, 1), blk, 0, stream>>>(x2, out2);
    // C: 32767 quad-threads/row -> 128 blocks
    wvlt_synthesis<<<dim3(128, B, 1), blk, 0, stream>>>(x2, w, out2, out3);
}